// Net_30614526886205
// MI455X (gfx1250) — compile-verified
//
#include <hip/hip_runtime.h>
#include <hip/hip_bf16.h>
#include <math.h>
#include <stdint.h>

// ---------------- problem constants (match reference) ----------------
constexpr int N_NODES = 1000000;
constexpr int E_EDGES = 16000000;
constexpr int K1 = 100000, K2 = 10000, K3 = 2500;
constexpr int NP1 = 1 << 20;   // pad for sort over N_NODES
constexpr int NP2 = 1 << 17;   // pad for sort over K1
constexpr int NP3 = 1 << 14;   // pad for sort over K2
constexpr float EPS = 1e-5f;

typedef __attribute__((ext_vector_type(2))) float v2f;
typedef __attribute__((ext_vector_type(8))) float v8f;

#define DEV __device__ __forceinline__

static inline int grid1(int n, int b = 256) { return (n + b - 1) / b; }

// ---------------- small device helpers ----------------
DEV float sigm_f(float t)     { return 1.0f / (1.0f + expf(-t)); }
DEV float softplus_f(float t) { return fmaxf(t, 0.0f) + log1pf(expf(-fabsf(t))); }

DEV unsigned fkey32(float f) {
  unsigned u = __float_as_uint(f);
  return (u & 0x80000000u) ? ~u : (u | 0x80000000u);   // monotone float->uint
}

// ---------------- scale scalars + accumulator zeroing ----------------
// scal[0..2] = sum(w)/||w|| for pw1/pw2/pw3 ; scal[4..11] = BN accumulators (zeroed)
__global__ void k_scales(const float* __restrict__ w1, const float* __restrict__ w2,
                         const float* __restrict__ w3, float* __restrict__ scal) {
  __shared__ float ss[256], sq[256];
  int t = threadIdx.x;
  for (int a = 0; a < 3; ++a) {
    const float* w = (a == 0) ? w1 : (a == 1) ? w2 : w3;
    int n = (a == 0) ? 800 : (a == 1) ? 80 : 8;
    float s = 0.f, q = 0.f;
    for (int i = t; i < n; i += 256) { float v = w[i]; s += v; q += v * v; }
    ss[t] = s; sq[t] = q; __syncthreads();
    for (int st = 128; st > 0; st >>= 1) {
      if (t < st) { ss[t] += ss[t + st]; sq[t] += sq[t + st]; }
      __syncthreads();
    }
    if (t == 0) scal[a] = ss[0] / sqrtf(sq[0]);
    __syncthreads();
  }
  if (t < 8) scal[4 + t] = 0.0f;
}

__global__ void k_zero_f32(float* __restrict__ p, int n) {
  int i = blockIdx.x * blockDim.x + threadIdx.x;
  if (i < n) p[i] = 0.0f;
}
__global__ void k_fill_i32(int* __restrict__ p, int n, int v) {
  int i = blockIdx.x * blockDim.x + threadIdx.x;
  if (i < n) p[i] = v;
}

// ---------------- CGConv edge kernels ----------------
__global__ void k_edge1(const float* __restrict__ x, const float* __restrict__ ea,
                        const int* __restrict__ src, const int* __restrict__ dst,
                        const float* __restrict__ Wf, const float* __restrict__ bf,
                        const float* __restrict__ Ws, const float* __restrict__ bs,
                        float* __restrict__ aggr, int E) {
  int e = blockIdx.x * blockDim.x + threadIdx.x;
  if (e >= E) return;
  int s = src[e], d = dst[e];
  float xi = x[d], xj = x[s], a = ea[e];                 // z = [x_dst, x_src, ea]
  float f = xi * Wf[0] + xj * Wf[1] + a * Wf[2] + bf[0];
  float g = xi * Ws[0] + xj * Ws[1] + a * Ws[2] + bs[0];
  atomicAdd(&aggr[d], sigm_f(f) * softplus_f(g));
}

__global__ void k_edge2(const float* __restrict__ x, const float* __restrict__ ea,
                        const int* __restrict__ src, const int* __restrict__ dst,
                        const int* __restrict__ remap,
                        const float* __restrict__ Wf, const float* __restrict__ bf,
                        const float* __restrict__ Ws, const float* __restrict__ bs,
                        float* __restrict__ aggr, int E) {
  int e = blockIdx.x * blockDim.x + threadIdx.x;
  if (e >= E) return;
  int s = remap[src[e]], d = remap[dst[e]];
  if (s < 0 || d < 0) return;                            // edge_valid mask
  float xi = x[d], xj = x[s], a = ea[e];
  float f = xi * Wf[0] + xj * Wf[1] + a * Wf[2] + bf[0];
  float g = xi * Ws[0] + xj * Ws[1] + a * Ws[2] + bs[0];
  atomicAdd(&aggr[d], sigm_f(f) * softplus_f(g));
}

// ---------------- batch-norm stats + apply ----------------
__global__ void k_stats(const float* __restrict__ a, int n,
                        float* __restrict__ sum, float* __restrict__ sumsq) {
  __shared__ float ss[256], sq[256];
  float s = 0.f, q = 0.f;
  for (int i = blockIdx.x * blockDim.x + threadIdx.x; i < n; i += gridDim.x * blockDim.x) {
    float v = a[i]; s += v; q += v * v;
  }
  ss[threadIdx.x] = s; sq[threadIdx.x] = q; __syncthreads();
  for (int st = 128; st > 0; st >>= 1) {
    if (threadIdx.x < st) { ss[threadIdx.x] += ss[threadIdx.x + st]; sq[threadIdx.x] += sq[threadIdx.x + st]; }
    __syncthreads();
  }
  if (threadIdx.x == 0) { atomicAdd(sum, ss[0]); atomicAdd(sumsq, sq[0]); }
}

__global__ void k_finstats(float* __restrict__ scal, int base, float n) {
  float mean = scal[base] / n;
  float var  = scal[base + 1] / n - mean * mean;         // biased var (matches jnp.var)
  scal[base + 2] = mean;
  scal[base + 3] = rsqrtf(var + EPS);
}

__global__ void k_bn_res(const float* __restrict__ xin, const float* __restrict__ aggr,
                         const float* __restrict__ scal, int base,
                         const float* __restrict__ g, const float* __restrict__ be,
                         float* __restrict__ xout, int n) {
  int i = blockIdx.x * blockDim.x + threadIdx.x;
  if (i >= n) return;
  xout[i] = g[0] * (aggr[i] - scal[base + 2]) * scal[base + 3] + be[0] + xin[i];
}

// ---------------- top-k: key build, bitonic sort (desc), extract ----------------
__global__ void k_keys(const float* __restrict__ x, int n,
                       const float* __restrict__ scal, int sidx,
                       unsigned long long* __restrict__ keys, int npad) {
  int i = blockIdx.x * blockDim.x + threadIdx.x;
  if (i >= npad) return;
  unsigned long long kk = 0ull;                          // padding sinks to tail
  if (i < n) {
    float sc = tanhf(x[i] * scal[sidx]);
    kk = ((unsigned long long)fkey32(sc) << 32)
       | (unsigned long long)(0xFFFFFFFFu - (unsigned)i); // stable: lower idx first
  }
  keys[i] = kk;
}

__global__ void k_bitonic(unsigned long long* __restrict__ keys, int j, int k, int npad) {
  int i = blockIdx.x * blockDim.x + threadIdx.x;
  if (i >= npad) return;
  int ix = i ^ j;
  if (ix <= i) return;
  unsigned long long a = keys[i], b = keys[ix];
  bool up = (i & k) == 0;                                // descending overall
  if (up ? (a < b) : (a > b)) { keys[i] = b; keys[ix] = a; }
}

__global__ void k_pool(const unsigned long long* __restrict__ keys,
                       const float* __restrict__ xin,
                       const float* __restrict__ scal, int sidx,
                       float* __restrict__ xout, int K) {
  int j = blockIdx.x * blockDim.x + threadIdx.x;
  if (j >= K) return;
  unsigned idx = 0xFFFFFFFFu - (unsigned)(keys[j] & 0xFFFFFFFFull);
  float xv = xin[idx];
  xout[j] = xv * tanhf(xv * scal[sidx]);                 // x[perm] * top_scores
}

__global__ void k_remap(const unsigned long long* __restrict__ keys,
                        int* __restrict__ remap, int K) {
  int j = blockIdx.x * blockDim.x + threadIdx.x;
  if (j >= K) return;
  unsigned idx = 0xFFFFFFFFu - (unsigned)(keys[j] & 0xFFFFFFFFull);
  remap[idx] = j;
}

// ---------------- final MLP via V_WMMA_F32_16X16X4_F32 + log_softmax ----------------
// One wave handles a 16-row tile of x5. h(16x8) = A(16x4:K0=x) * B(4x16:K0=Wl1) + bias
// Second layer (8->2) = butterfly reduction over cols in the WMMA D layout.
__global__ void k_mlp(const float* __restrict__ x,
                      const float* __restrict__ Wl1, const float* __restrict__ bl1,
                      const float* __restrict__ Wl2, const float* __restrict__ bl2,
                      float* __restrict__ out, int K) {
  int lane = threadIdx.x & 31;
  int wave = threadIdx.x >> 5;
  int tile = blockIdx.x * (blockDim.x >> 5) + wave;
  int row0 = tile * 16;

  // A operand: lanes 0-15 hold rows M=0..15 at K=0 (VGPR0); everything else 0
  float av = 0.0f;
  if (lane < 16) { int r = row0 + lane; if (r < K) av = x[r]; }
  v2f a = { av, 0.0f };
  // B operand: K=0 row, N=lane (lanes 0-15); only N<8 used
  float bv = (lane < 8) ? Wl1[lane] : 0.0f;
  v2f b = { bv, 0.0f };
  // C operand: bias bl1 broadcast to every row (D layout: col = lane&15)
  int col = lane & 15;
  float cb = (col < 8) ? bl1[col] : 0.0f;
  v8f c;
#pragma unroll
  for (int r = 0; r < 8; ++r) c[r] = cb;

  v8f h = __builtin_amdgcn_wmma_f32_16x16x4_f32(false, a, false, b, (short)0, c, false, false);

#pragma unroll
  for (int r = 0; r < 8; ++r) h[r] = fmaxf(h[r], 0.0f);  // relu

  // layer 2: o[m][ci] = sum_c h[m][c]*Wl2[c][ci] + bl2[ci]
  float w0 = (col < 8) ? Wl2[col * 2 + 0] : 0.0f;
  float w1 = (col < 8) ? Wl2[col * 2 + 1] : 0.0f;
  float b20 = bl2[0], b21 = bl2[1];
  float o0[8], o1[8];
#pragma unroll
  for (int r = 0; r < 8; ++r) {
    float t0 = h[r] * w0, t1 = h[r] * w1;
#pragma unroll
    for (int m = 1; m < 8; m <<= 1) {                    // reduce within groups of 8 lanes
      t0 += __shfl_xor(t0, m, 32);
      t1 += __shfl_xor(t1, m, 32);
    }
    o0[r] = t0 + b20; o1[r] = t1 + b21;
  }
  // lane 0 owns rows row0..row0+7 ; lane 16 owns rows row0+8..row0+15
  if (lane == 0 || lane == 16) {
    int rbase = row0 + ((lane == 16) ? 8 : 0);
#pragma unroll
    for (int r = 0; r < 8; ++r) {
      int row = rbase + r;
      if (row < K) {
        float p0 = o0[r], p1 = o1[r];
        float mx = fmaxf(p0, p1);
        float lse = mx + logf(expf(p0 - mx) + expf(p1 - mx));
        out[row * 2 + 0] = p0 - lse;
        out[row * 2 + 1] = p1 - lse;
      }
    }
  }
}

// ---------------- host-side helpers ----------------
static void bitonic_sort(unsigned long long* keys, int npad, hipStream_t stream) {
  int blocks = grid1(npad);
  for (int k = 2; k <= npad; k <<= 1)
    for (int j = k >> 1; j > 0; j >>= 1)
      k_bitonic<<<blocks, 256, 0, stream>>>(keys, j, k, npad);
}

extern "C" void kernel_launch(void* const* d_in, const int* in_sizes, int n_in,
                              void* d_out, int out_size, void* d_ws, size_t ws_size,
                              hipStream_t stream) {
  (void)in_sizes; (void)n_in; (void)out_size; (void)ws_size;
  // inputs in setup_inputs() dict order
  const float* x    = (const float*)d_in[0];
  const float* ea   = (const float*)d_in[1];
  const int*   src  = (const int*)d_in[2];
  const int*   dst  = (const int*)d_in[3];
  // d_in[4] = batch (unused: single graph)
  const float *Wf1 = (const float*)d_in[5],  *bf1 = (const float*)d_in[6];
  const float *Ws1 = (const float*)d_in[7],  *bs1 = (const float*)d_in[8];
  const float *g1  = (const float*)d_in[9],  *be1 = (const float*)d_in[10];
  const float *Wf2 = (const float*)d_in[11], *bf2 = (const float*)d_in[12];
  const float *Ws2 = (const float*)d_in[13], *bs2 = (const float*)d_in[14];
  const float *g2  = (const float*)d_in[15], *be2 = (const float*)d_in[16];
  const float *pw1 = (const float*)d_in[17], *pw2 = (const float*)d_in[18];
  const float *pw3 = (const float*)d_in[19];
  const float *Wl1 = (const float*)d_in[20], *bl1 = (const float*)d_in[21];
  const float *Wl2 = (const float*)d_in[22], *bl2 = (const float*)d_in[23];
  float* out = (float*)d_out;

  // workspace layout (~22 MB; everything hot fits in the 192 MB L2)
  char* ws = (char*)d_ws;
  float* scal  = (float*)ws;                                 // 64 floats
  float* aggr1 = (float*)(ws + 256);                         // N
  float* x1    = aggr1 + N_NODES;                            // N
  unsigned long long* keys = (unsigned long long*)(x1 + N_NODES); // NP1 (8B-aligned)
  int*   remap = (int*)(keys + NP1);                         // N
  float* x2    = (float*)(remap + N_NODES);                  // K1
  float* aggr2 = x2 + K1;                                    // K1
  float* x3    = aggr2 + K1;                                 // K1
  float* x4    = x3 + K1;                                    // K2
  float* x5    = x4 + K2;                                    // K3

  // ---- scoring scalars + zero BN accumulators ----
  k_scales<<<1, 256, 0, stream>>>(pw1, pw2, pw3, scal);

  // ---- CGConv #1 on full graph ----
  k_zero_f32<<<grid1(N_NODES), 256, 0, stream>>>(aggr1, N_NODES);
  k_edge1<<<grid1(E_EDGES), 256, 0, stream>>>(x, ea, src, dst, Wf1, bf1, Ws1, bs1, aggr1, E_EDGES);
  k_stats<<<1024, 256, 0, stream>>>(aggr1, N_NODES, &scal[4], &scal[5]);
  k_finstats<<<1, 1, 0, stream>>>(scal, 4, (float)N_NODES);
  k_bn_res<<<grid1(N_NODES), 256, 0, stream>>>(x, aggr1, scal, 4, g1, be1, x1, N_NODES);

  // ---- topk_pool #1 (N -> K1), exact sorted order ----
  k_keys<<<grid1(NP1), 256, 0, stream>>>(x1, N_NODES, scal, 0, keys, NP1);
  bitonic_sort(keys, NP1, stream);
  k_pool<<<grid1(K1), 256, 0, stream>>>(keys, x1, scal, 0, x2, K1);
  k_fill_i32<<<grid1(N_NODES), 256, 0, stream>>>(remap, N_NODES, -1);
  k_remap<<<grid1(K1), 256, 0, stream>>>(keys, remap, K1);

  // ---- CGConv #2 on pooled graph (edges remapped/masked on the fly) ----
  k_zero_f32<<<grid1(K1), 256, 0, stream>>>(aggr2, K1);
  k_edge2<<<grid1(E_EDGES), 256, 0, stream>>>(x2, ea, src, dst, remap, Wf2, bf2, Ws2, bs2, aggr2, E_EDGES);
  k_stats<<<256, 256, 0, stream>>>(aggr2, K1, &scal[8], &scal[9]);
  k_finstats<<<1, 1, 0, stream>>>(scal, 8, (float)K1);
  k_bn_res<<<grid1(K1), 256, 0, stream>>>(x2, aggr2, scal, 8, g2, be2, x3, K1);

  // ---- topk_pool #2 (K1 -> K2): edges are dead downstream, only x needed ----
  k_keys<<<grid1(NP2), 256, 0, stream>>>(x3, K1, scal, 1, keys, NP2);
  bitonic_sort(keys, NP2, stream);
  k_pool<<<grid1(K2), 256, 0, stream>>>(keys, x3, scal, 1, x4, K2);

  // ---- topk_pool #3 (K2 -> K3) ----
  k_keys<<<grid1(NP3), 256, 0, stream>>>(x4, K2, scal, 2, keys, NP3);
  bitonic_sort(keys, NP3, stream);
  k_pool<<<grid1(K3), 256, 0, stream>>>(keys, x4, scal, 2, x5, K3);

  // ---- MLP + log_softmax via WMMA (80 tiles of 16 rows: 10 blocks, 8 waves/block) ----
  k_mlp<<<grid1(K3 + 15, 256) + 0, 256, 0, stream>>>(x5, Wl1, bl1, Wl2, bl2, out, K3);
}